// GridPoolingLayer_26336739459657
// MI455X (gfx1250) — compile-verified
//
#include <hip/hip_runtime.h>
#include <stdint.h>

// ---------------------------------------------------------------------------
// GridPoolingLayer for MI455X (gfx1250, wave32).
// Memory-bound op (302 MB HBM floor ~13us @23.3TB/s). Cell-sum table (38MB)
// kept L2-resident (192MB L2). CDNA5 paths: TDM tensor_load_to_lds for input
// staging, s_wait_tensorcnt, v_wmma_f32_16x16x4_f32 one-hot gather.
// Workspace assumption: ws_size >= ~38.1 MB (32KB header + 385*385*64 floats).
// ---------------------------------------------------------------------------

#define H 768
#define W 768
#define C 64
#define NCMAX 385                 // max segment ids per axis: 384 rising edges + 1
#define NCELLS (NCMAX * NCMAX)    // 148225
#define ROWS_PER_BLK 4
#define COLS_PER_CHUNK 48
#define CHUNKS (W / COLS_PER_CHUNK)  // 16

typedef unsigned int u32x4 __attribute__((ext_vector_type(4)));
typedef int          i32x8 __attribute__((ext_vector_type(8)));
typedef int          i32x4 __attribute__((ext_vector_type(4)));
typedef float        v2f   __attribute__((ext_vector_type(2)));
typedef float        v8f   __attribute__((ext_vector_type(8)));

// ---- 1. segment ids + run lengths (serial, trivial: 2x768 iterations) -----
__global__ void build_segments(const int* __restrict__ hm, const int* __restrict__ vm,
                               int* __restrict__ row_ids, int* __restrict__ col_ids,
                               int* __restrict__ rowlen, int* __restrict__ collen,
                               int* __restrict__ meta) {
#if defined(__HIP_DEVICE_COMPILE__)
  for (int i = threadIdx.x; i < 512; i += blockDim.x) { rowlen[i] = 0; collen[i] = 0; }
  __syncthreads();
  if (threadIdx.x == 0) {
    int id = 0; row_ids[0] = 0; rowlen[0] = 1;
    for (int i = 1; i < H; ++i) {
      if (hm[i] == 1 && hm[i - 1] == 0) ++id;
      row_ids[i] = id; rowlen[id] += 1;
    }
    meta[0] = id + 1;
  } else if (threadIdx.x == 1) {
    int id = 0; col_ids[0] = 0; collen[0] = 1;
    for (int i = 1; i < W; ++i) {
      if (vm[i] == 1 && vm[i - 1] == 0) ++id;
      col_ids[i] = id; collen[id] += 1;
    }
    meta[1] = id + 1;
  }
#endif
}

// ---- 2. zero the L2-resident cell-sum table -------------------------------
__global__ void zero_sums(float4* __restrict__ p, long n4) {
#if defined(__HIP_DEVICE_COMPILE__)
  for (long i = (long)blockIdx.x * blockDim.x + threadIdx.x; i < n4;
       i += (long)gridDim.x * blockDim.x)
    p[i] = make_float4(0.f, 0.f, 0.f, 0.f);
#endif
}

// ---- 3. scatter: TDM tile -> LDS, per-row register scan, boundary atomics --
__global__ __launch_bounds__(128) void grid_scatter(const float* __restrict__ x,
                                                    const int* __restrict__ row_ids,
                                                    const int* __restrict__ col_ids,
                                                    float* __restrict__ sums) {
#if defined(__HIP_DEVICE_COMPILE__)
  __shared__ float tile[ROWS_PER_BLK * COLS_PER_CHUNK * C];  // 48 KB
  const int cb = blockIdx.x % CHUNKS;
  const int rb = blockIdx.x / CHUNKS;
  const int y0 = rb * ROWS_PER_BLK;
  const int w0 = cb * COLS_PER_CHUNK;

  if (threadIdx.x < 32) {  // wave 0 issues the DMA (wave-level op, uniform args)
    uint64_t ga  = (uint64_t)(uintptr_t)(x + ((size_t)y0 * W + w0) * C);
    uint32_t lds = (uint32_t)(uintptr_t)(&tile[0]);  // addr[31:0] == LDS offset (aperture rule)
    // D# group0: count=1, lds_addr, global_addr[56:0], type=2 (ISA 8.3)
    u32x4 g0;
    g0.x = 1u;
    g0.y = lds;
    g0.z = (uint32_t)ga;
    g0.w = ((uint32_t)(ga >> 32) & 0x01FFFFFFu) | 0x80000000u;
    // D# group1: data_size=4B; tensor_dim0=W*C, dim1=H; tile_dim0=48*64,
    // tile_dim1=4 rows; tensor_dim0_stride=W*C (row pitch in elements). (ISA 8.4)
    i32x8 g1;
    g1[0] = (int)(2u << 16);                                         // data_size=2 (4B)
    g1[1] = (int)(((uint32_t)(W * C) & 0xFFFFu) << 16);              // dim0[15:0] @ [31:16]
    g1[2] = (int)(((uint32_t)(W * C) >> 16) |
                  (((uint32_t)H & 0xFFFFu) << 16));                  // dim0[31:16] | dim1[15:0]
    g1[3] = (int)(((uint32_t)H >> 16) |
                  ((uint32_t)(COLS_PER_CHUNK * C) << 16));           // dim1[31:16] | tile_dim0
    g1[4] = ROWS_PER_BLK;                                            // tile_dim1 (tile_dim2=0)
    g1[5] = (int)(uint32_t)(W * C);                                  // dim0_stride[31:0]
    g1[6] = 0;
    g1[7] = 0;
    i32x4 zz4 = {0, 0, 0, 0};
    i32x8 zz8 = {0, 0, 0, 0, 0, 0, 0, 0};
    // clang-23 / therock 6-arg form: (g0, g1, g2, g3, g4, cpol).
    // 2-D non-cluster tensor: groups 2/3 (and the extra group) are zero-filled.
    __builtin_amdgcn_tensor_load_to_lds(g0, g1, zz4, zz4, zz8, 0);
    __builtin_amdgcn_s_wait_tensorcnt(0);
  }
  __syncthreads();

  const int wv = threadIdx.x >> 5;     // wave = row within block
  const int l  = threadIdx.x & 31;     // lane = channel pair {l, l+32}
  const int y  = y0 + wv;
  const int r  = row_ids[y];
  const float* trow = &tile[wv * COLS_PER_CHUNK * C];
  float a0 = 0.f, a1 = 0.f;
  int prev = col_ids[w0];
  for (int w = 0; w < COLS_PER_CHUNK; ++w) {
    int cid = col_ids[w0 + w];         // wave-uniform -> scalar branch, no divergence
    if (cid != prev) {
      float* dst = &sums[((size_t)r * NCMAX + prev) * C];
      atomicAdd(dst + l, a0);
      atomicAdd(dst + 32 + l, a1);
      a0 = 0.f; a1 = 0.f; prev = cid;
    }
    a0 += trow[w * C + l];
    a1 += trow[w * C + 32 + l];
  }
  float* dst = &sums[((size_t)r * NCMAX + prev) * C];
  atomicAdd(dst + l, a0);
  atomicAdd(dst + 32 + l, a1);
#endif
}

// ---- 4. sums -> means (area = rowlen*collen, exact, no atomic counts) -----
__global__ void cell_divide(float* __restrict__ sums, const int* __restrict__ rowlen,
                            const int* __restrict__ collen, const int* __restrict__ meta) {
#if defined(__HIP_DEVICE_COMPILE__)
  const int cell = blockIdx.x;
  const int r = cell / NCMAX, s = cell % NCMAX;
  if (r >= meta[0] || s >= meta[1]) return;
  const float area = (float)(rowlen[r] * collen[s]);
  const float inv = 1.0f / fmaxf(area, 1.0f);
  sums[(size_t)cell * C + threadIdx.x] *= inv;
#endif
}

// ---- 5. gather as one-hot WMMA: out(16px x 16ch) = G(16x12) * M(12x16) ----
// A-layout (ISA 7.12.2, f32 16x4): lane L holds A[L&15][kbase+ (L<16?0:2)], +1.
// B-layout: lane L holds B[(L<16?0:2)+..][L&15]. D: vgpr i -> M=i+(half*8), N=lane&15.
__global__ __launch_bounds__(128) void grid_gather(const float* __restrict__ means,
                                                   const int* __restrict__ row_ids,
                                                   const int* __restrict__ col_ids,
                                                   float* __restrict__ out) {
#if defined(__HIP_DEVICE_COMPILE__)
  const int tileId = blockIdx.x * 4 + (threadIdx.x >> 5);  // one wave per 16-px tile
  const int lane = threadIdx.x & 31;
  const int y  = tileId / (W / 16);
  const int w0 = (tileId % (W / 16)) * 16;
  const int r     = row_ids[y];       // uniform
  const int cbase = col_ids[w0];      // uniform
  const int half = lane >> 4;
  const int n    = lane & 15;
  const int lid  = col_ids[w0 + n] - cbase;  // local col-segment of pixel n (0..8 max)

  for (int cb = 0; cb < 4; ++cb) {
    v8f acc = {0.f, 0.f, 0.f, 0.f, 0.f, 0.f, 0.f, 0.f};
#pragma unroll
    for (int kk = 0; kk < 3; ++kk) {   // K window 12 covers lid<=8
      const int kb = kk * 4 + half * 2;
      v2f a;
      a.x = (lid == kb)     ? 1.0f : 0.0f;
      a.y = (lid == kb + 1) ? 1.0f : 0.0f;
      int s0 = cbase + kb;     s0 = s0 < NCMAX ? s0 : (NCMAX - 1);  // clamp: A is 0 there
      int s1 = cbase + kb + 1; s1 = s1 < NCMAX ? s1 : (NCMAX - 1);
      v2f b;
      b.x = means[((size_t)r * NCMAX + s0) * C + cb * 16 + n];
      b.y = means[((size_t)r * NCMAX + s1) * C + cb * 16 + n];
      acc = __builtin_amdgcn_wmma_f32_16x16x4_f32(false, a, false, b, (short)0, acc,
                                                  false, false);
    }
#pragma unroll
    for (int i = 0; i < 8; ++i) {
      const int M = i + half * 8;
      out[((size_t)y * W + w0 + M) * C + cb * 16 + n] = acc[i];
    }
  }
#endif
}

// ---------------------------------------------------------------------------
extern "C" void kernel_launch(void* const* d_in, const int* in_sizes, int n_in,
                              void* d_out, int out_size, void* d_ws, size_t ws_size,
                              hipStream_t stream) {
  const float* x  = (const float*)d_in[0];
  const int*   hm = (const int*)d_in[1];
  const int*   vm = (const int*)d_in[2];
  float* out = (float*)d_out;

  char* ws = (char*)d_ws;
  int*   row_ids = (int*)(ws + 0);
  int*   col_ids = (int*)(ws + 4096);
  int*   rowlen  = (int*)(ws + 8192);
  int*   collen  = (int*)(ws + 12288);
  int*   meta    = (int*)(ws + 16384);
  float* sums    = (float*)(ws + 32768);   // 148225 cells * 64 ch * 4 B = 37.9 MB

  hipLaunchKernelGGL(build_segments, dim3(1), dim3(256), 0, stream,
                     hm, vm, row_ids, col_ids, rowlen, collen, meta);

  const long n4 = (long)NCELLS * C / 4;
  hipLaunchKernelGGL(zero_sums, dim3(2048), dim3(256), 0, stream, (float4*)sums, n4);

  hipLaunchKernelGGL(grid_scatter, dim3((H / ROWS_PER_BLK) * CHUNKS), dim3(128), 0, stream,
                     x, row_ids, col_ids, sums);

  hipLaunchKernelGGL(cell_divide, dim3(NCELLS), dim3(C), 0, stream,
                     sums, rowlen, collen, meta);

  hipLaunchKernelGGL(grid_gather, dim3((H * (W / 16)) / 4), dim3(128), 0, stream,
                     sums, row_ids, col_ids, out);
}